// Net_24558622999235
// MI455X (gfx1250) — compile-verified
//
#include <hip/hip_runtime.h>
#include <math.h>

typedef __attribute__((ext_vector_type(2))) float v2f;
typedef __attribute__((ext_vector_type(8))) float v8f;

#define GCN_N     50000
#define GCN_E     1600000
#define GCN_K     2000
#define GCN_H     8

// ---------------------------------------------------------------- utilities
__global__ void k_zero(float* __restrict__ p, int n) {
    int i = blockIdx.x * blockDim.x + threadIdx.x;
    if (i < n) p[i] = 0.0f;
}

// deg[dst] += w  (real edges; self-loop weight 1 added in k_dinv)
__global__ void k_degree(const int* __restrict__ dst, const float* __restrict__ w,
                         float* __restrict__ deg, int e) {
    int i = blockIdx.x * blockDim.x + threadIdx.x;
    if (i < e) atomicAdd(&deg[dst[i]], w[i]);
}

// in-place: deg -> dinv = rsqrt(deg + 1)   (deg+1 > 0 always, matches jnp.where)
__global__ void k_dinv(float* __restrict__ deg, int n) {
    int i = blockIdx.x * blockDim.x + threadIdx.x;
    if (i < n) deg[i] = rsqrtf(deg[i] + 1.0f);
}

// norm[e] = dinv[src]*w*dinv[dst]  (real edges only)
__global__ void k_norm(const int* __restrict__ src, const int* __restrict__ dst,
                       const float* __restrict__ w, const float* __restrict__ dinv,
                       float* __restrict__ norm, int e) {
    int i = blockIdx.x * blockDim.x + threadIdx.x;
    if (i < e) norm[i] = dinv[src[i]] * w[i] * dinv[dst[i]];
}

// W1 [K x 8] row-major  ->  W1c [16 x K] column-major, cols 8..15 zero-padded
__global__ void k_w1_colmajor(const float* __restrict__ W1, float* __restrict__ W1c) {
    int i = blockIdx.x * blockDim.x + threadIdx.x;   // over 16*GCN_K
    if (i < 16 * GCN_K) {
        int n = i / GCN_K;     // output column 0..15
        int k = i % GCN_K;
        W1c[i] = (n < GCN_H) ? W1[k * GCN_H + n] : 0.0f;
    }
}

// ---------------------------------------------------------------- big GEMM:
// h1[N x 8] = x[N x 2000] @ W1 via V_WMMA_F32_16X16X4_F32, one wave per 16-row tile.
__global__ __launch_bounds__(256) void k_gemm1_wmma(const float* __restrict__ x,
                                                    const float* __restrict__ w1c,
                                                    float* __restrict__ h1) {
    const int wave = threadIdx.x >> 5;
    const int lane = threadIdx.x & 31;
    const int tile = blockIdx.x * 8 + wave;          // 16-row tile index
    if (tile >= GCN_N / 16) return;                  // wave-uniform: EXEC stays full

    const int half = lane >> 4;                      // 0: K slot 0..1, 1: K slot 2..3
    const int lm   = lane & 15;                      // A: M index / B: N index
    const int row  = tile * 16 + lm;

    const float* __restrict__ arow = x   + (size_t)row * GCN_K + 2 * half;
    const float* __restrict__ bcol = w1c + (size_t)lm  * GCN_K + 2 * half;

    v8f c = {};
#pragma unroll 4
    for (int kc = 0; kc < GCN_K / 4; ++kc) {
        v2f a = *(const v2f*)(arow + kc * 4);        // A[M=lm][k+2h .. k+2h+1]
        v2f b = *(const v2f*)(bcol + kc * 4);        // B[k+2h .. k+2h+1][N=lm]
        c = __builtin_amdgcn_wmma_f32_16x16x4_f32(
                /*neg_a=*/false, a, /*neg_b=*/false, b,
                /*c_mod=*/(short)0, c, /*reuse_a=*/false, /*reuse_b=*/false);
    }

    // C layout: VGPR v, lane -> (M = v + 8*half, N = lm). Keep only N < 8.
    if (lm < GCN_H) {
#pragma unroll
        for (int v = 0; v < 8; ++v) {
            int m = v + 8 * half;
            h1[(size_t)(tile * 16 + m) * GCN_H + lm] = c[v];
        }
    }
}

// agg[i][j] = bias[j] + dinv[i]^2 * h[i][j]   (self-loop contribution + bias)
__global__ void k_init_agg(const float* __restrict__ h, const float* __restrict__ dinv,
                           const float* __restrict__ bias, float* __restrict__ agg, int n) {
    int i = blockIdx.x * blockDim.x + threadIdx.x;   // over n*8
    if (i < n * GCN_H) {
        int node = i >> 3, j = i & 7;
        float di = dinv[node];
        agg[i] = bias[j] + di * di * h[i];
    }
}

// agg[dst] += norm[e] * h[src]   (8-wide rows, float4 gathers, f32 atomics)
__global__ void k_edge_scatter(const int* __restrict__ src, const int* __restrict__ dst,
                               const float* __restrict__ norm, const float* __restrict__ h,
                               float* __restrict__ agg, int e) {
    int i = blockIdx.x * blockDim.x + threadIdx.x;
    if (i >= e) return;
    int s = src[i], d = dst[i];
    float c = norm[i];
    const float4* hs = (const float4*)(h + (size_t)s * GCN_H);
    float4 h0 = hs[0], h1 = hs[1];
    float* out = agg + (size_t)d * GCN_H;
    atomicAdd(out + 0, c * h0.x);
    atomicAdd(out + 1, c * h0.y);
    atomicAdd(out + 2, c * h0.z);
    atomicAdd(out + 3, c * h0.w);
    atomicAdd(out + 4, c * h1.x);
    atomicAdd(out + 5, c * h1.y);
    atomicAdd(out + 6, c * h1.z);
    atomicAdd(out + 7, c * h1.w);
}

__global__ void k_relu(float* __restrict__ p, int n) {
    int i = blockIdx.x * blockDim.x + threadIdx.x;
    if (i < n) p[i] = fmaxf(p[i], 0.0f);
}

// h2[N x 8] = g1[N x 8] @ W2[8 x 8]  (tiny; one node per thread, W2 via cache)
__global__ void k_gemm2(const float* __restrict__ g1, const float* __restrict__ W2,
                        float* __restrict__ h2, int n) {
    int i = blockIdx.x * blockDim.x + threadIdx.x;
    if (i >= n) return;
    const float* g = g1 + (size_t)i * GCN_H;
    float acc[GCN_H];
#pragma unroll
    for (int j = 0; j < GCN_H; ++j) acc[j] = 0.0f;
#pragma unroll
    for (int k = 0; k < GCN_H; ++k) {
        float gk = g[k];
#pragma unroll
        for (int j = 0; j < GCN_H; ++j) acc[j] += gk * W2[k * GCN_H + j];
    }
    float* o = h2 + (size_t)i * GCN_H;
#pragma unroll
    for (int j = 0; j < GCN_H; ++j) o[j] = acc[j];
}

// in-place row log_softmax over 8 columns
__global__ void k_log_softmax(float* __restrict__ z, int n) {
    int i = blockIdx.x * blockDim.x + threadIdx.x;
    if (i >= n) return;
    float* r = z + (size_t)i * GCN_H;
    float v[GCN_H];
    float m = -INFINITY;
#pragma unroll
    for (int j = 0; j < GCN_H; ++j) { v[j] = r[j]; m = fmaxf(m, v[j]); }
    float s = 0.0f;
#pragma unroll
    for (int j = 0; j < GCN_H; ++j) s += expf(v[j] - m);
    float lse = m + logf(s);
#pragma unroll
    for (int j = 0; j < GCN_H; ++j) r[j] = v[j] - lse;
}

// ---------------------------------------------------------------- driver
extern "C" void kernel_launch(void* const* d_in, const int* in_sizes, int n_in,
                              void* d_out, int out_size, void* d_ws, size_t ws_size,
                              hipStream_t stream) {
    const float* x  = (const float*)d_in[0];
    const int*   src = (const int*)d_in[1];
    const int*   dst = (const int*)d_in[2];
    const float* ew  = (const float*)d_in[3];
    const float* W1  = (const float*)d_in[4];
    const float* b1  = (const float*)d_in[5];
    const float* W2  = (const float*)d_in[6];
    const float* b2  = (const float*)d_in[7];
    float* out = (float*)d_out;

    const int N = GCN_N, E = GCN_E;

    // workspace carve-up (floats)
    float* ws = (float*)d_ws;
    float* deg   = ws;                       // N   (becomes dinv in place)
    float* norm  = deg  + 50048;             // E
    float* w1c   = norm + GCN_E;             // 16*K = 32000
    float* h1    = w1c  + 16 * GCN_K;        // N*8
    float* agg1  = h1   + 400000;            // N*8  (relu'd in place -> g1)
    float* h2    = agg1 + 400000;            // N*8

    const int B = 256;
    auto g = [&](int n) { return (n + B - 1) / B; };

    // normalization
    k_zero<<<g(N), B, 0, stream>>>(deg, N);
    k_degree<<<g(E), B, 0, stream>>>(dst, ew, deg, E);
    k_dinv<<<g(N), B, 0, stream>>>(deg, N);                       // deg -> dinv
    k_norm<<<g(E), B, 0, stream>>>(src, dst, ew, deg, norm, E);

    // layer 1: h1 = x @ W1 (WMMA), agg = b1 + selfloop + edge scatter, relu
    k_w1_colmajor<<<g(16 * GCN_K), B, 0, stream>>>(W1, w1c);
    k_gemm1_wmma<<<(GCN_N / 16 + 7) / 8, 256, 0, stream>>>(x, w1c, h1);
    k_init_agg<<<g(N * GCN_H), B, 0, stream>>>(h1, deg, b1, agg1, N);
    k_edge_scatter<<<g(E), B, 0, stream>>>(src, dst, norm, h1, agg1, E);
    k_relu<<<g(N * GCN_H), B, 0, stream>>>(agg1, N * GCN_H);

    // layer 2: h2 = g1 @ W2, agg (into d_out), log_softmax in place
    k_gemm2<<<g(N), B, 0, stream>>>(agg1, W2, h2, N);
    k_init_agg<<<g(N * GCN_H), B, 0, stream>>>(h2, deg, b2, out, N);
    k_edge_scatter<<<g(E), B, 0, stream>>>(src, dst, norm, h2, out, E);
    k_log_softmax<<<g(N), B, 0, stream>>>(out, N);

    (void)in_sizes; (void)n_in; (void)out_size; (void)ws_size;
}